// PostProcess_22136261443789
// MI455X (gfx1250) — compile-verified
//
#include <hip/hip_runtime.h>
#include <hip/hip_bf16.h>
#include <math.h>

// ---------------- problem constants ----------------
#define NB      16          // batch
#define NA      25200       // anchors
#define NC      80          // classes
#define KCAND   2048        // PRE_NMS_TOPK
#define CAP     4096        // collection capacity before sort
#define NDET    300         // detections per image
#define SCORE_T 0.25f
#define NMS_T   0.45f
#define CLS_OFF 8192.0f     // per-class box offset (separates classes for NMS)
#define ABLK    256         // anchors per block in streaming passes
#define NBLKA   ((NA + ABLK - 1) / ABLK)

typedef __attribute__((ext_vector_type(2))) float v2f;
typedef __attribute__((ext_vector_type(8))) float v8f;

__device__ __forceinline__ float fast_rcp(float x) { return __builtin_amdgcn_rcpf(x); }
__device__ __forceinline__ float sgm(float x) { return fast_rcp(1.0f + __expf(-x)); }
__device__ __forceinline__ float fmin2(float a, float b) { return a < b ? a : b; }
__device__ __forceinline__ float fmax2(float a, float b) { return a > b ? a : b; }

// ---------------- workspace zero ----------------
__global__ void k_zero(unsigned* __restrict__ p, size_t n) {
    for (size_t i = (size_t)blockIdx.x * blockDim.x + threadIdx.x; i < n;
         i += (size_t)gridDim.x * blockDim.x)
        p[i] = 0u;
}

// ---- coalesced LDS staging of a 256-anchor x 85-float tile -----------------
__device__ __forceinline__ int stage_tile(const float* __restrict__ head,
                                          float* __restrict__ tile) {
    int b    = blockIdx.x / NBLKA;
    int a0   = (blockIdx.x % NBLKA) * ABLK;
    int cnt  = NA - a0; if (cnt > ABLK) cnt = ABLK;
    const float* g = head + ((size_t)b * NA + a0) * 85;
    int nf = cnt * 85;
    if (threadIdx.x == 0) __builtin_prefetch(g + ABLK * 85, 0, 0);  // global_prefetch_b8
    if (cnt == ABLK) {                                   // full tile: b128 coalesced
        const float4* g4 = (const float4*)g;
        float4* t4 = (float4*)tile;
        for (int k = threadIdx.x; k < (ABLK * 85) / 4; k += blockDim.x) t4[k] = g4[k];
    } else {
        for (int k = threadIdx.x; k < nf; k += blockDim.x) tile[k] = g[k];
    }
    __syncthreads();
    return cnt;
}

// per-anchor class-logit threshold: sig(c)*obj > cs  <=>  c > log(r/(1-r)), r=cs*(1+e)
__device__ __forceinline__ bool logit_thresh(float h4, float cs, float& obj, float& thrL) {
    float e = __expf(-h4);
    obj = fast_rcp(1.0f + e);
    float r = cs * (1.0f + e);
    if (r >= 1.0f) return false;          // even sig(c)=1 cannot pass
    thrL = __logf(r * fast_rcp(1.0f - r));
    return true;
}

// ---------------- pass 1: score histogram (streaming, memory-bound) -------------
__global__ void k_hist(const float* __restrict__ head, unsigned* __restrict__ hist) {
    __shared__ float tile[ABLK * 85];
    int cnt = stage_tile(head, tile);
    int b = blockIdx.x / NBLKA;
    int t = threadIdx.x;
    if (t >= cnt) return;
    const float* h = &tile[t * 85];
    float obj, thrL;
    if (!logit_thresh(h[4], SCORE_T, obj, thrL)) return;
    unsigned* hb = hist + (size_t)b * 2048;
    #pragma unroll 4
    for (int c = 0; c < NC; ++c) {
        float lc = h[5 + c];
        if (lc > thrL) {                  // rare: only then pay exp+rcp
            float s = sgm(lc) * obj;
            if (s > SCORE_T) {
                int bin = (int)(s * 2048.0f);
                bin = bin > 2047 ? 2047 : bin;
                atomicAdd(&hb[bin], 1u);
            }
        }
    }
}

// ---------------- pass 2a: find per-image score cutoff bin -----------------
__global__ void k_cut(const unsigned* __restrict__ hist, unsigned* __restrict__ cutbin) {
    int b = blockIdx.x;
    if (threadIdx.x != 0) return;
    const unsigned* hb = hist + (size_t)b * 2048;
    unsigned acc = 0;
    int cut = 512;                              // bin of SCORE_T: include everything
    for (int bin = 2047; bin >= 512; --bin) {
        acc += hb[bin];
        if (acc >= KCAND) { cut = bin; break; }
    }
    cutbin[b] = (unsigned)cut;
}

// ---------------- pass 2b: threshold-collect candidates --------------------
__global__ void k_collect(const float* __restrict__ head,
                          const unsigned* __restrict__ cutbin,
                          unsigned* __restrict__ ccnt,
                          float* __restrict__ cscore, unsigned* __restrict__ cidx) {
    __shared__ float tile[ABLK * 85];
    int cnt = stage_tile(head, tile);
    int b  = blockIdx.x / NBLKA;
    int a0 = (blockIdx.x % NBLKA) * ABLK;
    int t  = threadIdx.x;
    if (t >= cnt) return;
    int a = a0 + t;
    const float* h = &tile[t * 85];
    int   cut = (int)cutbin[b];
    float cs  = fmax2(SCORE_T, (float)cut * (1.0f / 2048.0f));
    float obj, thrL;
    if (!logit_thresh(h[4], cs, obj, thrL)) return;
    for (int c = 0; c < NC; ++c) {
        float lc = h[5 + c];
        if (lc > thrL) {
            float s = sgm(lc) * obj;
            int bin = (int)(s * 2048.0f);
            bin = bin > 2047 ? 2047 : bin;
            if (s > SCORE_T && bin >= cut) {
                unsigned pos = atomicAdd(&ccnt[b], 1u);
                if (pos < CAP) {
                    cscore[(size_t)b * CAP + pos] = s;
                    cidx[(size_t)b * CAP + pos]   = (unsigned)(a * NC + c);
                }
            }
        }
    }
}

// ---------------- pass 2c: per-image bitonic sort (LDS) + box decode -------
__global__ void k_sort(const float* __restrict__ head,
                       const float* __restrict__ grid,
                       const float* __restrict__ agrid,
                       const float* __restrict__ strd,
                       const unsigned* __restrict__ ccnt,
                       const float* __restrict__ cscore, const unsigned* __restrict__ cidx,
                       float* __restrict__ sscore, unsigned* __restrict__ slabel,
                       float* __restrict__ sbox, float* __restrict__ sobox) {
    __shared__ float    ss[CAP];
    __shared__ unsigned si[CAP];
    int b  = blockIdx.x;
    int bs = blockDim.x;
    unsigned n = ccnt[b]; if (n > CAP) n = CAP;
    for (int t = threadIdx.x; t < CAP; t += bs) {
        if ((unsigned)t < n) { ss[t] = cscore[(size_t)b * CAP + t]; si[t] = cidx[(size_t)b * CAP + t]; }
        else                 { ss[t] = -1.0f;                       si[t] = 0u; }
    }
    __syncthreads();
    // bitonic sort, descending by score
    for (int k = 2; k <= CAP; k <<= 1) {
        for (int j = k >> 1; j > 0; j >>= 1) {
            for (int i = threadIdx.x; i < CAP; i += bs) {
                int l = i ^ j;
                if (l > i) {
                    bool up = ((i & k) == 0);
                    float a0 = ss[i], a1 = ss[l];
                    if ((up && a0 < a1) || (!up && a0 > a1)) {
                        ss[i] = a1; ss[l] = a0;
                        unsigned t0 = si[i]; si[i] = si[l]; si[l] = t0;
                    }
                }
            }
            __syncthreads();
        }
    }
    // emit top-KCAND sorted candidates with decoded boxes
    for (int t = threadIdx.x; t < KCAND; t += bs) {
        size_t g = (size_t)b * KCAND + t;
        float s = ss[t];
        if (s > 0.0f) {
            unsigned id = si[t];
            int a = (int)(id / NC), c = (int)(id % NC);
            const float* h = head + ((size_t)b * NA + a) * 85;
            float px = sgm(h[0]), py = sgm(h[1]), pw = sgm(h[2]), ph = sgm(h[3]);
            float st = strd[a];
            float x  = (px * 2.0f - 0.5f + grid[a * 2 + 0]) * st;
            float y  = (py * 2.0f - 0.5f + grid[a * 2 + 1]) * st;
            float ww = (pw * 2.0f) * (pw * 2.0f) * agrid[a * 2 + 0];
            float hh = (ph * 2.0f) * (ph * 2.0f) * agrid[a * 2 + 1];
            float x0 = x - ww * 0.5f, y0 = y - hh * 0.5f;
            float x1 = x + ww * 0.5f, y1 = y + hh * 0.5f;
            float off = (float)c * CLS_OFF;
            sscore[g] = s; slabel[g] = (unsigned)c;
            sbox [g*4+0] = x0;       sbox [g*4+1] = y0;       sbox [g*4+2] = x1;       sbox [g*4+3] = y1;
            sobox[g*4+0] = x0 + off; sobox[g*4+1] = y0 + off; sobox[g*4+2] = x1 + off; sobox[g*4+3] = y1 + off;
        } else {
            sscore[g] = -1.0f; slabel[g] = 0xFFFFFFFFu;
            sbox [g*4+0] = 0; sbox [g*4+1] = 0; sbox [g*4+2] = 0; sbox [g*4+3] = 0;
            sobox[g*4+0] = 0; sobox[g*4+1] = 0; sobox[g*4+2] = 0; sobox[g*4+3] = 0;
        }
    }
}

// ---------------- pass 3: suppression bitmask (symmetric: ti<=tj only) ---------
// One wave32 per 16x16 (i,j) tile. WMMA computes area[i]+area[j] tile:
// A = [area_i | 1 | 0 | 0] (16x4), B = [1 ; area_j ; 0 ; 0] (4x16).
// IoU test division-free: inter > NMS_T * union. Transposed tile emitted from
// per-lane column bits (one shfl merges half-columns) -> ~2x less tile work.
__global__ void k_mask(const float* __restrict__ sobox,
                       const float* __restrict__ sscore,
                       unsigned* __restrict__ mask) {
    int wid  = threadIdx.x >> 5;
    int lane = threadIdx.x & 31;
    long tile = (long)blockIdx.x * (blockDim.x >> 5) + wid;
    int b  = (int)(tile / (128 * 128));
    int r  = (int)(tile % (128 * 128));
    int ti = r / 128, tj = r % 128;
    if (ti > tj) return;                                  // symmetry: lower half skipped
    const float* sc = sscore + (size_t)b * KCAND;
    if (sc[ti * 16] <= 0.0f || sc[tj * 16] <= 0.0f) return;   // sorted: whole tile empty
    const float4* ob = (const float4*)(sobox + (size_t)b * KCAND * 4);

    int jg = tj * 16 + (lane & 15);
    float4 bj = ob[jg];
    float areaj = (bj.z - bj.x) * (bj.w - bj.y);

    int il = ti * 16 + (lane & 15);
    float4 bi0 = ob[il];
    float areai = (bi0.z - bi0.x) * (bi0.w - bi0.y);

    bool lo = (lane < 16);
    v2f a2, b2;
    a2.x = lo ? areai : 0.0f;   // A K=0 row values (lanes 0-15), K=2 zero (lanes 16-31)
    a2.y = lo ? 1.0f  : 0.0f;   // A K=1 ones,                    K=3 zero
    b2.x = lo ? 1.0f  : 0.0f;   // B row0 = ones, row2 = zero
    b2.y = lo ? areaj : 0.0f;   // B row1 = area_j, row3 = zero
    v8f acc = {};
    acc = __builtin_amdgcn_wmma_f32_16x16x4_f32(false, a2, false, b2,
                                                (short)0, acc, false, false);
    int half = (lane >> 4) << 3;    // lanes 16-31 own rows v+8
    unsigned colbits = 0;           // this lane's column (n = lane&15), rows v(+half)
    unsigned shj = (unsigned)(tj & 1) * 16u;
    #pragma unroll
    for (int v = 0; v < 8; ++v) {
        int ig = ti * 16 + v + half;
        float4 bi = ob[ig];
        float ltx = fmax2(bi.x, bj.x), lty = fmax2(bi.y, bj.y);
        float rbx = fmin2(bi.z, bj.z), rby = fmin2(bi.w, bj.w);
        float iw = fmax2(rbx - ltx, 0.0f), ih = fmax2(rby - lty, 0.0f);
        float inter = iw * ih;
        float uni = acc[v] - inter + 1e-7f;          // union via WMMA area sum
        bool bit = (inter > NMS_T * uni) && (ig != jg);
        colbits |= bit ? (1u << v) : 0u;
        unsigned bal = (unsigned)__ballot(bit);      // [15:0]=row v, [31:16]=row v+8
        if (lane == 0) {
            atomicOr(&mask[((size_t)b * KCAND + ti*16 + v    ) * 64 + (tj >> 1)],
                     (bal & 0xFFFFu) << shj);
            atomicOr(&mask[((size_t)b * KCAND + ti*16 + v + 8) * 64 + (tj >> 1)],
                     (bal >> 16) << shj);
        }
    }
    // transposed tile (tj,ti): column n of (ti,tj) becomes row tj*16+n
    unsigned partner = __shfl(colbits, lane ^ 16);   // other half-column (rows +8)
    if (ti != tj && lane < 16) {
        unsigned full = colbits | (partner << 8);    // 16 rows of column lane
        unsigned shi = (unsigned)(ti & 1) * 16u;
        atomicOr(&mask[((size_t)b * KCAND + tj*16 + lane) * 64 + (ti >> 1)],
                 full << shi);
    }
}

// ---------------- pass 4: greedy bit-reduce, single wave32, all-register --------
__global__ void k_reduce(const float* __restrict__ sscore,
                         const unsigned* __restrict__ mask,
                         unsigned* __restrict__ keep) {
    int b = blockIdx.x, lane = threadIdx.x;        // 32 threads = 1 wave, no barriers
    const float* sc = sscore + (size_t)b * KCAND;
    unsigned kp0 = 0, kp1 = 0;                     // keep words lane, lane+32
    for (int bit = 0; bit < 32; ++bit) {
        if (sc[lane * 32 + bit]        > 0.0f) kp0 |= (1u << bit);
        if (sc[(lane + 32) * 32 + bit] > 0.0f) kp1 |= (1u << bit);
    }
    const unsigned* mb = mask + (size_t)b * KCAND * 64;
    for (int i = 0; i < KCAND; ++i) {
        __builtin_prefetch(&mb[(size_t)(i + 8) * 64 + lane * 2], 0, 0); // pull row ahead
        int wi = i >> 5;
        unsigned cand = (wi < 32) ? kp0 : kp1;
        unsigned word = __shfl(cand, wi & 31);
        if ((word >> (i & 31)) & 1u) {
            kp0 &= ~mb[(size_t)i * 64 + lane];
            kp1 &= ~mb[(size_t)i * 64 + 32 + lane];
        }
    }
    keep[b * 64 + lane]      = kp0;
    keep[b * 64 + 32 + lane] = kp1;
}

// ---------------- pass 5: parallel rank + scatter of first 300 kept ------------
__global__ void k_final(const unsigned* __restrict__ keep,
                        const float* __restrict__ sscore,
                        const unsigned* __restrict__ slabel,
                        const float* __restrict__ sbox,
                        float* __restrict__ out) {
    __shared__ unsigned pc[64];
    int b = blockIdx.x, t = threadIdx.x;           // 64 threads
    float* obx = out;                              // [NB][NDET][4]
    float* osc = out + (size_t)NB * NDET * 4;      // [NB][NDET]
    float* olb = osc + (size_t)NB * NDET;          // [NB][NDET]
    unsigned w = keep[b * 64 + t];
    pc[t] = __popc(w);
    __syncthreads();
    unsigned p = 0;
    for (int k = 0; k < t; ++k) p += pc[k];        // exclusive prefix (64 words, cheap)
    unsigned total = p; for (int k = t; k < 64; ++k) total += pc[k];
    // scatter this word's kept bits at ranks p, p+1, ...
    unsigned r = p;
    for (int bit = 0; bit < 32; ++bit) {
        if ((w >> bit) & 1u) {
            if (r < NDET) {
                size_t g = (size_t)b * KCAND + t * 32 + bit;
                size_t o = (size_t)b * NDET + r;
                obx[o*4+0] = sbox[g*4+0]; obx[o*4+1] = sbox[g*4+1];
                obx[o*4+2] = sbox[g*4+2]; obx[o*4+3] = sbox[g*4+3];
                osc[o] = sscore[g];
                olb[o] = (float)slabel[g];
            }
            ++r;
        }
    }
    // cooperative padding [total, NDET)
    int start = (int)(total < NDET ? total : NDET);
    for (int k = start + t; k < NDET; k += 64) {
        size_t o = (size_t)b * NDET + k;
        obx[o*4+0] = 0; obx[o*4+1] = 0; obx[o*4+2] = 0; obx[o*4+3] = 0;
        osc[o] = -1.0f;
        olb[o] = -1.0f;
    }
}

extern "C" void kernel_launch(void* const* d_in, const int* in_sizes, int n_in,
                              void* d_out, int out_size, void* d_ws, size_t ws_size,
                              hipStream_t stream) {
    const float* head  = (const float*)d_in[0];   // [16,25200,85]
    const float* grid  = (const float*)d_in[1];   // [25200,2]
    const float* agrid = (const float*)d_in[2];   // [25200,2]
    const float* strd  = (const float*)d_in[3];   // [25200,1]
    float* out = (float*)d_out;
    char* w = (char*)d_ws;

    size_t o = 0;
    auto alloc = [&](size_t bytes) { size_t r = o; o += (bytes + 255) & ~(size_t)255; return r; };
    size_t oHist = alloc((size_t)NB * 2048 * sizeof(unsigned));
    size_t oCut  = alloc((size_t)NB * sizeof(unsigned));
    size_t oCnt  = alloc((size_t)NB * sizeof(unsigned));
    size_t oCS   = alloc((size_t)NB * CAP * sizeof(float));
    size_t oCI   = alloc((size_t)NB * CAP * sizeof(unsigned));
    size_t oSS   = alloc((size_t)NB * KCAND * sizeof(float));
    size_t oSL   = alloc((size_t)NB * KCAND * sizeof(unsigned));
    size_t oSB   = alloc((size_t)NB * KCAND * 4 * sizeof(float));
    size_t oSO   = alloc((size_t)NB * KCAND * 4 * sizeof(float));
    size_t oMask = alloc((size_t)NB * KCAND * 64 * sizeof(unsigned));
    size_t oKeep = alloc((size_t)NB * 64 * sizeof(unsigned));
    size_t total = o;

    unsigned* hist   = (unsigned*)(w + oHist);
    unsigned* cutbin = (unsigned*)(w + oCut);
    unsigned* ccnt   = (unsigned*)(w + oCnt);
    float*    cscore = (float*)   (w + oCS);
    unsigned* cidx   = (unsigned*)(w + oCI);
    float*    sscore = (float*)   (w + oSS);
    unsigned* slabel = (unsigned*)(w + oSL);
    float*    sbox   = (float*)   (w + oSB);
    float*    sobox  = (float*)   (w + oSO);
    unsigned* maskp  = (unsigned*)(w + oMask);
    unsigned* keepp  = (unsigned*)(w + oKeep);

    k_zero<<<2048, 256, 0, stream>>>((unsigned*)w, total / 4);

    int sblk = NB * NBLKA;                 // streaming passes: 256 anchors / block
    k_hist<<<sblk, ABLK, 0, stream>>>(head, hist);
    k_cut<<<NB, 32, 0, stream>>>(hist, cutbin);
    k_collect<<<sblk, ABLK, 0, stream>>>(head, cutbin, ccnt, cscore, cidx);
    k_sort<<<NB, 512, 0, stream>>>(head, grid, agrid, strd, ccnt, cscore, cidx,
                                   sscore, slabel, sbox, sobox);
    // NB * 128 * 128 tiles, 8 waves (tiles) per 256-thread block (ti>tj exit early)
    int mblk = (NB * 128 * 128) / 8;
    k_mask<<<mblk, 256, 0, stream>>>(sobox, sscore, maskp);
    k_reduce<<<NB, 32, 0, stream>>>(sscore, maskp, keepp);
    k_final<<<NB, 64, 0, stream>>>(keepp, sscore, slabel, sbox, out);
}